// network_712964571382
// MI455X (gfx1250) — compile-verified
//
#include <hip/hip_runtime.h>

#define NN 100000      // N_NODES  (divisible by 16)
#define NE 1200000     // N_EDGES
#define NG 2048        // N_GRAPHS (divisible by 16)
#define ND 74          // NODE_DIM
#define NH 64          // HIDDEN
#define NL 4           // N_GCN

#define LDSA  68       // padded LDS row stride (floats) for K=64 A tiles
#define LDSAE 76       // padded LDS row stride (floats) for embed (K=74 -> 76)

typedef __attribute__((ext_vector_type(2))) float v2f;
typedef __attribute__((ext_vector_type(8))) float v8f;

// ---------------- Tensor Data Mover (TDM) helpers ----------------
#if __has_builtin(__builtin_amdgcn_tensor_load_to_lds) && __has_builtin(__builtin_amdgcn_s_wait_tensorcnt)
#define HAVE_TDM 1
typedef unsigned int v4u __attribute__((ext_vector_type(4)));
typedef int          v8i __attribute__((ext_vector_type(8)));
typedef int          v4i __attribute__((ext_vector_type(4)));

// 2D tile load: tile_d0 x tile_d1 elements (4B each), global row stride stride0,
// optional LDS padding of 4 dwords after every 64 dwords stored (row stride 64 -> 68).
__device__ __forceinline__ void tdm_load(unsigned lds_off, const void* gptr,
                                         unsigned tile_d0, unsigned tile_d1,
                                         unsigned tensor_d0, unsigned tensor_d1,
                                         unsigned long long stride0, bool pad64to68)
{
    const unsigned long long ga = (unsigned long long)gptr;
    v4u g0;
    g0[0] = 1u;                                        // count=1, user mode
    g0[1] = lds_off;                                   // lds_addr [63:32]
    g0[2] = (unsigned)(ga & 0xFFFFFFFFu);              // global_addr [95:64]
    g0[3] = (unsigned)((ga >> 32) & 0x1FFFFFFu) | (2u << 30); // addr[56:32] | type=2

    unsigned d0 = (2u << 16);                          // data_size = 4 bytes
    if (pad64to68) d0 |= (1u << 20) | (5u << 22) | (3u << 25); // pad_en, every 64 dw, +4 dw
    v8i g1;
    g1[0] = (int)d0;
    g1[1] = (int)((tensor_d0 & 0xFFFFu) << 16);                        // dim0[15:0] @ 63:48
    g1[2] = (int)(((tensor_d0 >> 16) & 0xFFFFu) | ((tensor_d1 & 0xFFFFu) << 16));
    g1[3] = (int)(((tensor_d1 >> 16) & 0xFFFFu) | ((tile_d0 & 0xFFFFu) << 16));
    g1[4] = (int)(tile_d1 & 0xFFFFu);                                  // tile_dim1, tile_dim2=0
    g1[5] = (int)(stride0 & 0xFFFFFFFFull);                            // dim0_stride[31:0]
    g1[6] = (int)((stride0 >> 32) & 0xFFFFull);                        // dim0_stride[47:32]
    g1[7] = 0;
    const v4i z4 = {0, 0, 0, 0};
    const v8i z8 = {0, 0, 0, 0, 0, 0, 0, 0};
    __builtin_amdgcn_tensor_load_to_lds(g0, g1, z4, z4, z8, 0);   // 6-arg form
}

__device__ __forceinline__ unsigned lds_off_of(const void* p)
{
    return (unsigned)(unsigned long long)p;            // aperture: LDS offset is the low 32 bits of the address
}
#else
#define HAVE_TDM 0
#endif

// ---------------- WMMA inner loop from LDS ----------------
// acc = sA[0:16, 0:Kp] (row stride ldsa, per-row scale s) @ sW[0:Kp, n0:n0+16]  (ldb = 64)
// V_WMMA_F32_16X16X4_F32 fragments: A V0=K|K+2, V1=K+1|K+3; B mirrored.
__device__ __forceinline__ v8f wmma_loop(const float* __restrict__ sA, int ldsa,
                                         const float* __restrict__ sW,
                                         int Kp, int n0, float s)
{
    const int lane = threadIdx.x & 31;
    const int hi   = lane >> 4;
    const int lr   = lane & 15;
    v8f acc = {0.f, 0.f, 0.f, 0.f, 0.f, 0.f, 0.f, 0.f};
    #pragma unroll
    for (int k0 = 0; k0 < Kp; k0 += 4) {
        const int ka = k0 + hi * 2;
        v2f a, b;
        a.x = sA[lr * ldsa + ka]     * s;   // ds_load_b64 (consecutive pair)
        a.y = sA[lr * ldsa + ka + 1] * s;
        b.x = sW[ka       * NH + n0 + lr];
        b.y = sW[(ka + 1) * NH + n0 + lr];
        acc = __builtin_amdgcn_wmma_f32_16x16x4_f32(false, a, false, b,
                                                    (short)0, acc, false, false);
    }
    return acc;
}

// C/D layout: lane holds col n0+(lane&15); VGPR v holds row row0 + v + 8*(lane>>4).
__device__ __forceinline__ void store_tile(float* __restrict__ out, int row0, int n0,
                                           v8f acc, const float* __restrict__ bias,
                                           const float* __restrict__ resid, bool do_relu)
{
    const int lane = threadIdx.x & 31;
    const int hi   = lane >> 4;
    const int n    = n0 + (lane & 15);
    const float bv = bias ? bias[n] : 0.f;
    #pragma unroll
    for (int v = 0; v < 8; ++v) {
        const int r = row0 + v + 8 * hi;
        float x = acc[v] + bv;
        if (do_relu) x = fmaxf(x, 0.f);
        if (resid) x += resid[(long)r * NH + n];
        out[(long)r * NH + n] = x;
    }
}

// ---------------- kernels ----------------

__global__ void k_degrees(const int* __restrict__ ei,
                          float* __restrict__ deg_out, float* __restrict__ deg_in)
{
    const int i = blockIdx.x * blockDim.x + threadIdx.x;
    if (i < NE) {
        atomicAdd(&deg_out[ei[i]], 1.f);
        atomicAdd(&deg_in[ei[NE + i]], 1.f);
    }
}

__global__ void k_norms(float* __restrict__ a, float* __restrict__ b)
{
    const int i = blockIdx.x * blockDim.x + threadIdx.x;
    if (i < NN) {
        const float d0 = a[i];
        a[i] = d0 > 0.f ? rsqrtf(fmaxf(d0, 1.f)) : 0.f;
        const float d1 = b[i];
        b[i] = d1 > 0.f ? rsqrtf(fmaxf(d1, 1.f)) : 0.f;
    }
}

// h = node_feats @ W_embed + b_embed   (K = 74, zero-padded to 76 in LDS)
__global__ __launch_bounds__(128) void k_embed(const float* __restrict__ X,
                                               const float* __restrict__ W,
                                               const float* __restrict__ bias,
                                               float* __restrict__ h)
{
    __shared__ float sA[16 * LDSAE];
    __shared__ float sW[LDSAE * NH];
    const int tid  = threadIdx.x;
    const int row0 = blockIdx.x * 16;

    for (int i = tid; i < 16 * LDSAE; i += 128) {
        const int r = i / LDSAE, c = i % LDSAE;
        sA[i] = (c < ND) ? X[(long)(row0 + r) * ND + c] : 0.f;
    }
    for (int i = tid; i < LDSAE * NH; i += 128) {
        sW[i] = (i < ND * NH) ? W[i] : 0.f;
    }
    __syncthreads();

    const int n0 = (tid >> 5) * 16;
    v8f acc = wmma_loop(sA, LDSAE, sW, LDSAE, n0, 1.0f);
    store_tile(h, row0, n0, acc, bias, nullptr, false);
}

// agg[dst] += h[src] * norm_s[src]   (wave per edge; float2 per lane)
__global__ __launch_bounds__(256) void k_scatter(const int* __restrict__ ei,
                                                 const float* __restrict__ h,
                                                 const float* __restrict__ norm_s,
                                                 float* __restrict__ agg)
{
    const int gid  = blockIdx.x * blockDim.x + threadIdx.x;
    const int e    = gid >> 5;
    const int lane = threadIdx.x & 31;
    if (e >= NE) return;

    const int s    = ei[e];
    const int d    = ei[NE + e];
    const float ns = norm_s[s];

    if (lane == 0 && e + 64 < NE) {
        const int sp = ei[e + 64];
        __builtin_prefetch(h + (long)sp * NH, 0, 0);   // global_prefetch_b8
    }

    const float2 v = ((const float2*)(h + (long)s * NH))[lane];  // global_load_b64
    float* ad = agg + (long)d * NH + lane * 2;
    atomicAdd(ad,     v.x * ns);
    atomicAdd(ad + 1, v.y * ns);
}

// h = h + relu((agg * norm_d) @ W + b)   (K = 64; A tile + W staged via TDM)
__global__ __launch_bounds__(128) void k_update(const float* __restrict__ agg,
                                                const float* __restrict__ norm_d,
                                                const float* __restrict__ W,
                                                const float* __restrict__ bias,
                                                float* __restrict__ h)
{
    __shared__ float sA[16 * LDSA];
    __shared__ float sW[NH * NH];
    const int tid  = threadIdx.x;
    const int row0 = blockIdx.x * 16;

#if HAVE_TDM
    if (tid < 32) {          // wave 0 issues both TDM descriptors (EXEC ignored by TDM)
        tdm_load(lds_off_of(sA), agg + (long)row0 * NH, NH, 16, NH, 16, NH, true);
        tdm_load(lds_off_of(sW), W, NH * NH, 0, NH * NH, 1, NH * NH, false);
        __builtin_amdgcn_s_wait_tensorcnt(0);
    }
#else
    for (int i = tid; i < 16 * NH; i += 128) {
        const int r = i >> 6, c = i & 63;
        sA[r * LDSA + c] = agg[(long)(row0 + r) * NH + c];
    }
    for (int i = tid; i < NH * NH; i += 128) sW[i] = W[i];
#endif
    __syncthreads();

    const float s = norm_d[row0 + (tid & 15)];   // per-A-row scale
    const int n0  = (tid >> 5) * 16;
    v8f acc = wmma_loop(sA, LDSA, sW, NH, n0, s);
    store_tile(h, row0, n0, acc, bias, h, true);
}

// hg[node2graph[n]] += h[n]
__global__ void k_pool(const float* __restrict__ h, const int* __restrict__ n2g,
                       float* __restrict__ hg)
{
    const long i = (long)blockIdx.x * blockDim.x + threadIdx.x;
    if (i < (long)NN * NH) {
        const int n = (int)(i >> 6);
        const int c = (int)(i & 63);
        atomicAdd(&hg[(long)n2g[n] * NH + c], h[i]);
    }
}

// out = relu(hg @ W1 + b1) @ W2 + b2
__global__ __launch_bounds__(128) void k_head(const float* __restrict__ hg,
                                              const float* __restrict__ W1,
                                              const float* __restrict__ b1,
                                              const float* __restrict__ W2,
                                              const float* __restrict__ b2,
                                              float* __restrict__ out)
{
    __shared__ float sA[16 * LDSA];
    __shared__ float sW[NH * NH];
    __shared__ float t[16][NH + 1];
    const int tid  = threadIdx.x;
    const int lane = tid & 31;
    const int row0 = blockIdx.x * 16;

#if HAVE_TDM
    if (tid < 32) {
        tdm_load(lds_off_of(sA), hg + (long)row0 * NH, NH, 16, NH, 16, NH, true);
        tdm_load(lds_off_of(sW), W1, NH * NH, 0, NH * NH, 1, NH * NH, false);
        __builtin_amdgcn_s_wait_tensorcnt(0);
    }
#else
    for (int i = tid; i < 16 * NH; i += 128) {
        const int r = i >> 6, c = i & 63;
        sA[r * LDSA + c] = hg[(long)(row0 + r) * NH + c];
    }
    for (int i = tid; i < NH * NH; i += 128) sW[i] = W1[i];
#endif
    __syncthreads();

    const int n0 = (tid >> 5) * 16;
    v8f acc = wmma_loop(sA, LDSA, sW, NH, n0, 1.0f);

    const int hi = lane >> 4;
    const int n  = n0 + (lane & 15);
    #pragma unroll
    for (int v = 0; v < 8; ++v)
        t[v + 8 * hi][n] = fmaxf(acc[v] + b1[n], 0.f);
    __syncthreads();

    if (tid < 16) {
        float sum = 0.f;
        #pragma unroll
        for (int c = 0; c < NH; ++c) sum += t[tid][c] * W2[c];
        out[row0 + tid] = sum + b2[0];
    }
}

// ---------------- launcher ----------------

extern "C" void kernel_launch(void* const* d_in, const int* in_sizes, int n_in,
                              void* d_out, int out_size, void* d_ws, size_t ws_size,
                              hipStream_t stream)
{
    const float* node_feats = (const float*)d_in[0];
    const float* W_embed    = (const float*)d_in[1];
    const float* b_embed    = (const float*)d_in[2];
    const float* W_gcn      = (const float*)d_in[3];
    const float* b_gcn      = (const float*)d_in[4];
    const float* W_lin1     = (const float*)d_in[5];
    const float* b_lin1     = (const float*)d_in[6];
    const float* W_lin2     = (const float*)d_in[7];
    const float* b_lin2     = (const float*)d_in[8];
    const int*   edge_index = (const int*)d_in[9];
    const int*   node2graph = (const int*)d_in[10];
    float* out = (float*)d_out;

    float* deg_out = (float*)d_ws;                 // -> norm_s in place
    float* deg_in  = deg_out + NN;                 // -> norm_d in place
    float* h       = deg_in  + NN;                 // [NN, 64]
    float* agg     = h   + (size_t)NN * NH;        // [NN, 64]
    float* hg      = agg + (size_t)NN * NH;        // [NG, 64]

    (void)hipMemsetAsync(deg_out, 0, 2ull * NN * sizeof(float), stream);
    k_degrees<<<(NE + 255) / 256, 256, 0, stream>>>(edge_index, deg_out, deg_in);
    k_norms<<<(NN + 255) / 256, 256, 0, stream>>>(deg_out, deg_in);

    const int gemm_blocks = NN / 16;
    k_embed<<<gemm_blocks, 128, 0, stream>>>(node_feats, W_embed, b_embed, h);

    for (int l = 0; l < NL; ++l) {
        (void)hipMemsetAsync(agg, 0, (size_t)NN * NH * sizeof(float), stream);
        k_scatter<<<(int)(((long)NE * 32 + 255) / 256), 256, 0, stream>>>(edge_index, h, deg_out, agg);
        k_update<<<gemm_blocks, 128, 0, stream>>>(agg, deg_in,
                                                  W_gcn + (size_t)l * NH * NH,
                                                  b_gcn + (size_t)l * NH, h);
    }

    (void)hipMemsetAsync(hg, 0, (size_t)NG * NH * sizeof(float), stream);
    k_pool<<<(int)(((long)NN * NH + 255) / 256), 256, 0, stream>>>(h, node2graph, hg);
    k_head<<<NG / 16, 128, 0, stream>>>(hg, W_lin1, b_lin1, W_lin2, b_lin2, out);
}